// SubsetRouting_76699525972645
// MI455X (gfx1250) — compile-verified
//
#include <hip/hip_runtime.h>
#include <stdint.h>

typedef __attribute__((ext_vector_type(2))) float    v2f;
typedef __attribute__((ext_vector_type(8))) float    v8f;
typedef __attribute__((ext_vector_type(4))) uint32_t v4u;
typedef __attribute__((ext_vector_type(8))) uint32_t v8u;

#define I_CAPS   1152
#define O_CAPS   32
#define DDIM     16
#define KTH      922          // ceil(0.8 * 1152)
#define ROWPITCH 17           // 16 dwords of data + 1 pad dword (from TDM pad) -> bank-conflict-free
#define NTHREADS 256

// ---- dynamic LDS layout (bytes) ----
#define SU_BYTES    (I_CAPS * ROWPITCH * 4)          // 78336: padded u tile
#define SKEY_OFF    (SU_BYTES)
#define SKEY_BYTES  (I_CAPS * 4)                     // flipped loss keys
#define SNORM_OFF   (SKEY_OFF + SKEY_BYTES)
#define SNORM_BYTES (I_CAPS * 4)                     // per-row norms
#define BINS_OFF    (SNORM_OFF + SNORM_BYTES)
#define BINS_BYTES  (256 * 4)                        // radix-select histogram
#define ACC_OFF     (BINS_OFF + BINS_BYTES)
#define ACC_BYTES   (17 * 4)                         // num[16] + den
#define SEL_OFF     (ACC_OFF + ACC_BYTES)
#define SEL_BYTES   (2 * 4)                          // {prefix, k}
#define SMEM_TOTAL  (SEL_OFF + SEL_BYTES)            // ~88.7 KB < 320 KB/WGP

__global__ __launch_bounds__(NTHREADS)
void subset_routing_kernel(const float* __restrict__ u, float* __restrict__ out)
{
    extern __shared__ unsigned char smem[];
    float*    sU    = (float*)smem;
    uint32_t* sKey  = (uint32_t*)(smem + SKEY_OFF);
    float*    sNorm = (float*)(smem + SNORM_OFF);
    uint32_t* bins  = (uint32_t*)(smem + BINS_OFF);
    float*    sAcc  = (float*)(smem + ACC_OFF);
    uint32_t* sSel  = (uint32_t*)(smem + SEL_OFF);

    const int t    = threadIdx.x;
    const int lane = t & 31;
    const int wave = t >> 5;
    const int bo   = blockIdx.x;            // = b*O_CAPS + o

    // ================= Phase A: TDM load of u[b, :, o, :] into LDS =================
    // Tile: 1152 rows x 16 f32, row stride 512 elements in memory; LDS rows padded
    // to 17 dwords by the TDM (pad every 64 B by 1 dword).
    if (wave == 0) {   // wave-level branch: TDM ignores EXEC, so must be branched around
        const int b = bo >> 5, o = bo & 31;
        const uint64_t gaddr = (uint64_t)(uintptr_t)u
                             + ((((uint64_t)b * I_CAPS * O_CAPS) + (uint64_t)o) * DDIM) * 4ull;
        const uint32_t lds_off = (uint32_t)(uintptr_t)(void*)sU;   // flat addr low 32 bits == LDS offset

        v4u g0; v8u g1;
        // ---- D# group 0 (128b) ----
        g0.x = 1u;                                                  // count=1, user descriptor
        g0.y = lds_off;                                             // lds_addr [63:32]
        g0.z = (uint32_t)gaddr;                                     // global_addr [95:64]
        g0.w = ((uint32_t)(gaddr >> 32) & 0x01FFFFFFu) | 0x80000000u; // addr[56:32] | type=2
        // ---- D# group 1 (256b) ----
        g1.s0 = (2u << 16)      // data_size = 4 bytes
              | (1u << 20)      // pad_enable
              | (3u << 22);     // pad_interval = 3 -> every 16 dwords ; pad_amount=0 -> 1 dword
        g1.s1 = (uint32_t)DDIM   << 16;   // tensor_dim0 lo16 @ bits[63:48]
        g1.s2 = (uint32_t)I_CAPS << 16;   // tensor_dim0 hi=0; tensor_dim1 lo16 @ bits[95:80]
        g1.s3 = (uint32_t)DDIM   << 16;   // tensor_dim1 hi=0; tile_dim0 @ bits[127:112]
        g1.s4 = (uint32_t)I_CAPS;         // tile_dim1 @ [143:128]; tile_dim2=0
        g1.s5 = (uint32_t)(O_CAPS * DDIM);// tensor_dim0_stride = 512 elements (bits 191:160)
        g1.s6 = 0u;                       // stride hi; tensor_dim1_stride (unused, 2D)
        g1.s7 = 0u;
        asm volatile("tensor_load_to_lds %0, %1" :: "s"(g0), "s"(g1) : "memory");
        __builtin_amdgcn_s_wait_tensorcnt(0);
    }
    __syncthreads();

    // ================= Phase B: norms + weighted sums (first average) ==============
    if (t < 17) sAcc[t] = 0.0f;
    __syncthreads();
    {
        float myNum[DDIM]; float myDen = 0.0f;
        #pragma unroll
        for (int d = 0; d < DDIM; ++d) myNum[d] = 0.0f;
        for (int i = t; i < I_CAPS; i += NTHREADS) {
            const float* row = &sU[i * ROWPITCH];
            float s = 0.0f;
            #pragma unroll
            for (int d = 0; d < DDIM; ++d) { float x = row[d]; s += x * x; }
            float n = sqrtf(s);
            sNorm[i] = n;
            myDen += n;
            #pragma unroll
            for (int d = 0; d < DDIM; ++d) myNum[d] += row[d] * n;
        }
        #pragma unroll
        for (int d = 0; d < DDIM; ++d) atomicAdd(&sAcc[d], myNum[d]);
        atomicAdd(&sAcc[16], myDen);
    }
    __syncthreads();

    float v[DDIM];
    {
        const float invden = 1.0f / sAcc[16];
        #pragma unroll
        for (int d = 0; d < DDIM; ++d) v[d] = sAcc[d] * invden;
    }

    // ============ Phase C: losses[-<v,u_i>] via V_WMMA_F32_16X16X4_F32 =============
    // A (16x4 f32): lane<16 holds row=lane, K=0,1 in VGPR0,1; lane>=16 holds K=2,3.
    // B (4x16 f32): every column gets v-chunk -> each D column holds the row dot.
    {
        const int  mrow = lane & 15;
        const bool lo   = (lane < 16);
        for (int g = wave; g < I_CAPS / 16; g += NTHREADS / 32) {
            const float* rowp = &sU[(g * 16 + mrow) * ROWPITCH];
            v8f acc = {0.f, 0.f, 0.f, 0.f, 0.f, 0.f, 0.f, 0.f};
            #pragma unroll
            for (int c = 0; c < 4; ++c) {
                v2f a, bb;
                a.x  = lo ? rowp[4 * c + 0] : rowp[4 * c + 2];
                a.y  = lo ? rowp[4 * c + 1] : rowp[4 * c + 3];
                bb.x = lo ? v[4 * c + 0]    : v[4 * c + 2];
                bb.y = lo ? v[4 * c + 1]    : v[4 * c + 3];
                acc = __builtin_amdgcn_wmma_f32_16x16x4_f32(
                          false, a, false, bb, (short)0, acc, false, false);
            }
            // D layout: lanes 0-15 hold M=0..7 (VGPR 0..7), lanes 16-31 hold M=8..15.
            if ((lane & 15) == 0) {
                const int rbase = g * 16 + ((lane >> 4) << 3);
                #pragma unroll
                for (int j = 0; j < 8; ++j) {
                    float loss = -acc[j];
                    uint32_t kk = __float_as_uint(loss);
                    kk ^= (kk & 0x80000000u) ? 0xFFFFFFFFu : 0x80000000u; // order-preserving flip
                    sKey[rbase + j] = kk;
                }
            }
        }
    }
    if (t == 0) { sSel[0] = 0u; sSel[1] = KTH; }
    __syncthreads();

    // ============== Phase D: radix select kth-smallest key (k = 922) ===============
    for (int pass = 0; pass < 4; ++pass) {
        bins[t] = 0u;
        __syncthreads();
        const uint32_t prefix = sSel[0];
        const int shift = 24 - 8 * pass;
        for (int i = t; i < I_CAPS; i += NTHREADS) {
            const uint32_t key = sKey[i];
            if (((uint64_t)key >> (shift + 8)) == (uint64_t)prefix)
                atomicAdd(&bins[(key >> shift) & 0xFFu], 1u);
        }
        __syncthreads();
        if (t == 0) {
            uint32_t kk = sSel[1], cum = 0, b2 = 0;
            for (b2 = 0; b2 < 256; ++b2) {
                const uint32_t c2 = bins[b2];
                if (cum + c2 >= kk) break;
                cum += c2;
            }
            sSel[0] = (prefix << 8) | b2;
            sSel[1] = kk - cum;
        }
        __syncthreads();
    }
    const uint32_t threshKey = sSel[0];

    // ================= Phase E: masked weighted average + store ====================
    if (t < 17) sAcc[t] = 0.0f;
    __syncthreads();
    {
        float myNum[DDIM]; float myDen = 0.0f;
        #pragma unroll
        for (int d = 0; d < DDIM; ++d) myNum[d] = 0.0f;
        for (int i = t; i < I_CAPS; i += NTHREADS) {
            if (sKey[i] <= threshKey) {          // choose: loss <= kth-value (ties included)
                const float n = sNorm[i];        // ||u*choose|| = choose * ||u||
                const float* row = &sU[i * ROWPITCH];
                myDen += n;
                #pragma unroll
                for (int d = 0; d < DDIM; ++d) myNum[d] += row[d] * n;
            }
        }
        #pragma unroll
        for (int d = 0; d < DDIM; ++d) atomicAdd(&sAcc[d], myNum[d]);
        atomicAdd(&sAcc[16], myDen);
    }
    __syncthreads();

    if (t < DDIM)
        out[bo * DDIM + t] = sAcc[t] / sAcc[16];
}

extern "C" void kernel_launch(void* const* d_in, const int* in_sizes, int n_in,
                              void* d_out, int out_size, void* d_ws, size_t ws_size,
                              hipStream_t stream) {
    const float* u = (const float*)d_in[0];
    float* out = (float*)d_out;
    const int B = in_sizes[0] / (I_CAPS * O_CAPS * DDIM);   // 128
    dim3 grid((unsigned)(B * O_CAPS));                      // one block per (b, o) tile
    dim3 block(NTHREADS);
    hipLaunchKernelGGL(subset_routing_kernel, grid, block, SMEM_TOTAL, stream, u, out);
}